// Point2Mask_87686052315595
// MI455X (gfx1250) — compile-verified
//
#include <hip/hip_runtime.h>
#include <math.h>

// Point2Mask for MI455X (gfx1250, wave32).
//
// d2(s,p) = (sx-px)^2 + (sy-py)^2 via V_WMMA_F32_16X16X4_F32 (K=4), with
// POINTS on the M axis and SAMPLES on the N axis:
//   A row (point):   [-2px, -2py, px^2+py^2, 1 ]     (rebuilt per 16-pt tile)
//   B col (sample):  [ sx,   sy,  1,  sx^2+sy^2]     (loop-invariant)
// D layout (lane L, VGPR v -> M = v + 8*(L>=16), N = L%16) puts 8 consecutive
// points of one sample straight into the lane's accumulator registers, so the
// ordered ball-query (first <=16 points with d2<4, point-index order) runs
// from registers. Lanes L and L+16 co-own sample L%16 (points 0-7 / 8-15);
// the budget hand-off needs only one __shfl_xor(.,16) per tile because the
// consumed count is min(#within, budget) and #within is budget-independent.

typedef __attribute__((ext_vector_type(2))) float v2f;
typedef __attribute__((ext_vector_type(8))) float v8f;

#define KINST 8
#define MVIEW 4
#define NPER  1024
#define NFEAT 20
#define RES   48
#define NSAMP 16
#define RAD2  4.0f

__global__ __launch_bounds__(256) void p2m_kernel(const float* __restrict__ xyz,
                                                  const float* __restrict__ feats,
                                                  const float* __restrict__ theta,
                                                  const float* __restrict__ phi,
                                                  float* __restrict__ out)
{
    __shared__ __align__(16) float4 s_pt[NPER];   // (px, py, px^2+py^2, -)
    __shared__ __align__(16) float  s_f0[NPER];   // second-largest feature
    __shared__ __align__(16) float  s_f1[NPER];   // largest feature
    __shared__ float4 s_red[256];                 // min/max reduction scratch

    const int tid = threadIdx.x;
    const int b   = blockIdx.x;            // batch = k*MVIEW + m
    const int kin = b / MVIEW;
    const int m   = b % MVIEW;

    // ---- projection basis for view m ----
    const float th = theta[m], ph = phi[m];
    const float st = sinf(th), ct = cosf(th);
    const float sp = sinf(ph), cp = cosf(ph);
    const float Ux = -st,     Uy = ct;                    // Uz = 0
    const float Vx = ct * sp, Vy = st * sp, Vz = cp;
    const float cx = ct * cp, cy = st * cp, cz = sp;      // r = 1

    // ---- phase 1: raw projected coords + per-thread min/max ----
    float minu = 3.0e38f, maxu = -3.0e38f, minv = 3.0e38f, maxv = -3.0e38f;
    for (int j = tid; j < NPER; j += 256) {
        const int gn = kin * NPER + j;
        const float x = xyz[gn * 3 + 0] - cx;
        const float y = xyz[gn * 3 + 1] - cy;
        const float z = xyz[gn * 3 + 2] - cz;
        const float cu = x * Ux + y * Uy;
        const float cv = x * Vx + y * Vy + z * Vz;
        s_pt[j].x = cu; s_pt[j].y = cv;
        minu = fminf(minu, cu); maxu = fmaxf(maxu, cu);
        minv = fminf(minv, cv); maxv = fmaxf(maxv, cv);
    }
    s_red[tid] = make_float4(minu, maxu, minv, maxv);
    __syncthreads();
    for (int s = 128; s > 0; s >>= 1) {
        if (tid < s) {
            float4 a = s_red[tid], c = s_red[tid + s];
            a.x = fminf(a.x, c.x); a.y = fmaxf(a.y, c.y);
            a.z = fminf(a.z, c.z); a.w = fmaxf(a.w, c.w);
            s_red[tid] = a;
        }
        __syncthreads();
    }
    const float4 r0  = s_red[0];
    const float cenu = (r0.y + r0.x) * 0.5f;
    const float sclu = fmaxf(r0.y - r0.x, 1e-5f) * 0.5f;
    const float cenv = (r0.w + r0.z) * 0.5f;
    const float sclv = fmaxf(r0.w - r0.z, 1e-5f) * 0.5f;

    // ---- phase 2: normalize coords, precompute |p|^2 and top-2 features ----
    for (int j = tid; j < NPER; j += 256) {
        const float pu = ((s_pt[j].x - cenu) / sclu + 1.0f) * (0.8f * (float)RES * 0.5f)
                         + 0.1f * (float)RES;
        const float pv = ((s_pt[j].y - cenv) / sclv + 1.0f) * (0.8f * (float)RES * 0.5f)
                         + 0.1f * (float)RES;
        s_pt[j] = make_float4(pu, pv, pu * pu + pv * pv, 0.0f);

        const float* f = feats + (size_t)(kin * NPER + j) * NFEAT;
        float m1 = -3.4e38f, m2 = -3.4e38f;
        #pragma unroll
        for (int c = 0; c < NFEAT; ++c) {
            const float v = f[c];
            if (v > m1) { m2 = m1; m1 = v; } else if (v > m2) { m2 = v; }
        }
        s_f0[j] = m2; s_f1[j] = m1;
    }
    __syncthreads();

    // ---- phase 3: per-wave 32-sample tile, WMMA over 64 point-tiles ----
    const int lane = tid & 31;
    const int wave = tid >> 5;
    const int s0   = (blockIdx.y * 8 + wave) * 32;   // 9 * 8 * 32 = 2304 samples
    const bool loHalf = (lane < 16);
    const int  rbase  = loHalf ? 0 : 8;              // my 8 points within a tile

    // B fragments (samples, loop-invariant).
    // 4x16 f32 layout mirrors A: lanes 0-15 hold K0/K1, lanes 16-31 hold K2/K3.
    const int sLo = s0 + (lane & 15);
    const int sHi = sLo + 16;
    const float xLo = (float)(sLo / RES), yLo = (float)(sLo % RES);
    const float xHi = (float)(sHi / RES), yHi = (float)(sHi % RES);
    v2f bLo, bHi;
    bLo.x = loHalf ? xLo : 1.0f;                         // K0 | K2
    bLo.y = loHalf ? yLo : (xLo * xLo + yLo * yLo);      // K1 | K3
    bHi.x = loHalf ? xHi : 1.0f;
    bHi.y = loHalf ? yHi : (xHi * xHi + yHi * yHi);

    // two sample states per lane: A <- sample sLo (c0), B <- sample sHi (c1)
    int   remA = NSAMP, remB = NSAMP;
    float fsA0 = 0.0f, fsA1 = 0.0f, fsB0 = 0.0f, fsB1 = 0.0f;

    for (int t = 0; t < NPER / 16; ++t) {
        const int p0 = t * 16;

        // A fragment (points): lanes 0-15: (-2px,-2py); lanes 16-31: (pp, 1)
        const float4 pt = s_pt[p0 + (lane & 15)];        // one ds_load_b128
        v2f af;
        af.x = loHalf ? (-2.0f * pt.x) : pt.z;
        af.y = loHalf ? (-2.0f * pt.y) : 1.0f;

        v8f c0 = {}; v8f c1 = {};
        c0 = __builtin_amdgcn_wmma_f32_16x16x4_f32(false, af, false, bLo,
                                                   (short)0, c0, false, false);
        c1 = __builtin_amdgcn_wmma_f32_16x16x4_f32(false, af, false, bHi,
                                                   (short)0, c1, false, false);

        // features of MY 8 points (shared by both sample states)
        float gf0[8], gf1[8];
        *(float4*)&gf0[0] = *(const float4*)&s_f0[p0 + rbase];
        *(float4*)&gf0[4] = *(const float4*)&s_f0[p0 + rbase + 4];
        *(float4*)&gf1[0] = *(const float4*)&s_f1[p0 + rbase];
        *(float4*)&gf1[4] = *(const float4*)&s_f1[p0 + rbase + 4];

        // within flags + budget-independent counts
        int cntA = 0, cntB = 0;
        bool wA[8], wB[8];
        #pragma unroll
        for (int v = 0; v < 8; ++v) {
            wA[v] = c0[v] < RAD2; cntA += wA[v] ? 1 : 0;
            wB[v] = c1[v] < RAD2; cntB += wB[v] ? 1 : 0;
        }
        // exchange counts with the half-pair partner (lane ^ 16)
        const int packed = cntA | (cntB << 8);
        const int other  = __shfl_xor(packed, 16, 32);
        const int cAo = other & 0xff;
        const int cBo = (other >> 8) & 0xff;

        // my starting budget: low half goes first, high half starts after it
        int budA = loHalf ? remA : max(remA - cAo, 0);
        int budB = loHalf ? remB : max(remB - cBo, 0);

        // ordered scan of my 8 points, straight from WMMA accumulators
        #pragma unroll
        for (int v = 0; v < 8; ++v) {
            const bool sA = wA[v] && (budA > 0);
            fsA0 += sA ? gf0[v] : 0.0f;
            fsA1 += sA ? gf1[v] : 0.0f;
            budA -= sA ? 1 : 0;
            const bool sB = wB[v] && (budB > 0);
            fsB0 += sB ? gf0[v] : 0.0f;
            fsB1 += sB ? gf1[v] : 0.0f;
            budB -= sB ? 1 : 0;
        }
        // consistent budget update in both halves
        remA = max(remA - cntA - cAo, 0);
        remB = max(remB - cntB - cBo, 0);

        // all 32 samples of this wave saturated -> rest of the scan is a no-op
        if (!__any((remA | remB) > 0)) break;
    }

    // merge half-pair partial sums (selection order preserved; sum order not)
    fsA0 += __shfl_xor(fsA0, 16, 32);
    fsA1 += __shfl_xor(fsA1, 16, 32);
    fsB0 += __shfl_xor(fsB0, 16, 32);
    fsB1 += __shfl_xor(fsB1, 16, 32);

    // lane L outputs sample s0 + L: low half -> state A, high half -> state B
    const float fs0 = loHalf ? fsA0 : fsB0;
    const float fs1 = loHalf ? fsA1 : fsB1;
    const int   rem = loHalf ? remA : remB;

    // ---- epilogue: softmax over (second-max, max) feature means ----
    const float cnt = (float)(NSAMP - rem);
    const float occ = fmaxf(cnt, 1.0f);
    const float e0  = fs0 / occ;
    const float e1  = fs1 / occ;
    float val = 0.0f;
    if (e0 != e1) {
        const float mx = fmaxf(e0, e1);
        const float z0 = expf(e0 - mx), z1 = expf(e1 - mx);
        val = z1 / (z0 + z1);          // nf[...,1]; "empty" (equal) -> 0
    }
    val *= 255.0f;

    const int smp = s0 + lane;
    const int xg  = smp / RES, yg = smp % RES;
    const size_t base = ((size_t)b * 3) * RES * RES + (size_t)xg * RES + yg;
    out[base + 0 * RES * RES] = val;
    out[base + 1 * RES * RES] = val;
    out[base + 2 * RES * RES] = val;
}

extern "C" void kernel_launch(void* const* d_in, const int* in_sizes, int n_in,
                              void* d_out, int out_size, void* d_ws, size_t ws_size,
                              hipStream_t stream) {
    (void)in_sizes; (void)n_in; (void)d_ws; (void)ws_size; (void)out_size;
    const float* xyz   = (const float*)d_in[0];
    const float* feats = (const float*)d_in[1];
    // d_in[2] = proposals (contiguous equal blocks, implied by layout)
    // d_in[3] = res (== 48, compile-time constant here)
    const float* theta = (const float*)d_in[4];
    const float* phi   = (const float*)d_in[5];
    float* out = (float*)d_out;

    dim3 grid(KINST * MVIEW, (RES * RES) / 256);   // 32 batches x 9 sample groups
    p2m_kernel<<<grid, 256, 0, stream>>>(xyz, feats, theta, phi, out);
}